// spafusion_6433861009811
// MI455X (gfx1250) — compile-verified
//
#include <hip/hip_runtime.h>

#define LEPS 1e-5f

// ---------------------------------------------------------------------------
// Problem constants (B=2, C=64, H=W=48, NH=8, hd=8)
// ---------------------------------------------------------------------------
static constexpr int Bn = 2;
static constexpr int Cn = 64;
static constexpr int Hn = 48;
static constexpr int Wn = 48;
static constexpr int Nn = Hn * Wn;   // 2304 tokens per batch
static constexpr int Rn = Bn * Nn;   // 4608 total rows

typedef __attribute__((ext_vector_type(2))) float v2f;
typedef __attribute__((ext_vector_type(8))) float v8f;

// CDNA5 f32 WMMA: D(16x16,f32) = A(16x4,f32) x B(4x16,f32) + C
// 8-arg form: (neg_a, A, neg_b, B, c_mod, C, reuse_a, reuse_b)
__device__ __forceinline__ v8f wmma4(v2f a, v2f b, v8f c) {
  return __builtin_amdgcn_wmma_f32_16x16x4_f32(false, a, false, b, (short)0, c,
                                               false, false);
}

// ---- CDNA5 async global->LDS copy (ASYNCcnt path), guarded by probe --------
#if defined(__has_builtin)
#if __has_builtin(__builtin_amdgcn_global_load_async_to_lds_b128) && \
    __has_builtin(__builtin_amdgcn_s_wait_asynccnt)
#define USE_ASYNC_LDS 1
#endif
#endif

#ifdef USE_ASYNC_LDS
// builtin prototype (from hipcc diagnostic): pointers are int4 in AS(1)/AS(3)
typedef int gvec4i __attribute__((vector_size(4 * sizeof(int))));
typedef __attribute__((address_space(1))) gvec4i* as1_v4i_ptr;
typedef __attribute__((address_space(3))) gvec4i* as3_v4i_ptr;

__device__ __forceinline__ void async_b128(const float* g, float* l) {
  __builtin_amdgcn_global_load_async_to_lds_b128((as1_v4i_ptr)g, (as3_v4i_ptr)l,
                                                 0, 0);
}
#endif

// wave32 full reduction (sum)
__device__ __forceinline__ float wave_sum32(float x) {
#pragma unroll
  for (int m = 16; m >= 1; m >>= 1) x += __shfl_xor(x, m, 32);
  return x;
}
// reduction within each 16-lane half (rows of a WMMA C/D tile live per-half)
__device__ __forceinline__ float half_max16(float x) {
#pragma unroll
  for (int m = 8; m >= 1; m >>= 1) x = fmaxf(x, __shfl_xor(x, m, 32));
  return x;
}
__device__ __forceinline__ float half_sum16(float x) {
#pragma unroll
  for (int m = 8; m >= 1; m >>= 1) x += __shfl_xor(x, m, 32);
  return x;
}

// ---------------------------------------------------------------------------
// 1) NCHW -> row-major [B*N, 2C] pack (x | y), enables coalesced LN + GEMM A
// ---------------------------------------------------------------------------
__global__ void __launch_bounds__(256) pack_kernel(const float* __restrict__ x,
                                                   const float* __restrict__ y,
                                                   float* __restrict__ X2) {
  int idx = blockIdx.x * 256 + threadIdx.x;  // over Rn*128
  int c = idx & 127;
  int row = idx >> 7;
  int b = row / Nn, p = row % Nn;
  X2[idx] = (c < 64) ? x[(b * Cn + c) * Nn + p]
                     : y[(b * Cn + (c - 64)) * Nn + p];
}

// ---------------------------------------------------------------------------
// 2) Generic WMMA GEMM: out[M,Nc] = A[M,K] @ W[Nc,K]^T + bias   (row-major)
//    one wave = one 16x16 output tile
// ---------------------------------------------------------------------------
__global__ void __launch_bounds__(256) gemm_rm_kernel(
    const float* __restrict__ A, const float* __restrict__ W,
    const float* __restrict__ bias, float* __restrict__ out, int K, int Nc) {
  const int wid = blockIdx.x * blockDim.y + threadIdx.y;
  const int tilesN = Nc >> 4;
  const int mt = wid / tilesN, nt = wid % tilesN;
  const int lane = threadIdx.x;
  const int r = lane & 15;
  const int kh = (lane >> 4) << 1;
  const int hi8 = (lane >> 4) << 3;
  const float* Ar = A + (size_t)(mt * 16 + r) * K;
  const float* Wr = W + (size_t)(nt * 16 + r) * K;
  v8f acc = {};
  for (int k0 = 0; k0 < K; k0 += 4) {
    v2f a, b;
    a[0] = Ar[k0 + kh];     a[1] = Ar[k0 + kh + 1];
    b[0] = Wr[k0 + kh];     b[1] = Wr[k0 + kh + 1];
    acc = wmma4(a, b, acc);
  }
  int col = nt * 16 + r;
  float bv = bias ? bias[col] : 0.f;
#pragma unroll
  for (int i = 0; i < 8; ++i)
    out[(size_t)(mt * 16 + i + hi8) * Nc + col] = acc[i] + bv;
}

// ---------------------------------------------------------------------------
// 3) fused BatchNorm (eval) + 3x LayerNorm; wave per row
// ---------------------------------------------------------------------------
__global__ void __launch_bounds__(256) bnln_kernel(
    const float* __restrict__ X2, const float* __restrict__ XY,
    const float* bg, const float* bb_, const float* bm, const float* bv,
    const float* lxg, const float* lxb, const float* lyg, const float* lyb,
    const float* lzg, const float* lzb, float* __restrict__ sc1,
    float* __restrict__ sd1, float* __restrict__ xl, float* __restrict__ yl,
    float* __restrict__ zl) {
  const int row = blockIdx.x * blockDim.y + threadIdx.y;
  const int lane = threadIdx.x;
  const int c0 = lane, c1 = lane + 32;

  float s0 = bg[c0] * rsqrtf(bv[c0] + LEPS);
  float s1 = bg[c1] * rsqrtf(bv[c1] + LEPS);
  float sh0 = bb_[c0] - bm[c0] * s0, sh1 = bb_[c1] - bm[c1] * s1;

  float x0 = X2[row * 128 + c0] * s0 + sh0;
  float x1 = X2[row * 128 + c1] * s1 + sh1;
  float y0 = X2[row * 128 + 64 + c0] * s0 + sh0;
  float y1 = X2[row * 128 + 64 + c1] * s1 + sh1;
  float z0 = XY[row * 64 + c0] * s0 + sh0;
  float z1 = XY[row * 64 + c1] * s1 + sh1;

  sc1[row * 64 + c0] = x0; sc1[row * 64 + c1] = x1;
  sd1[row * 64 + c0] = y0; sd1[row * 64 + c1] = y1;

  auto LN = [&](float a0, float a1, const float* g, const float* be, float* o) {
    float mu = wave_sum32(a0 + a1) * (1.f / 64.f);
    float d0 = a0 - mu, d1 = a1 - mu;
    float var = wave_sum32(d0 * d0 + d1 * d1) * (1.f / 64.f);
    float inv = rsqrtf(var + LEPS);
    o[row * 64 + c0] = d0 * inv * g[c0] + be[c0];
    o[row * 64 + c1] = d1 * inv * g[c1] + be[c1];
  };
  LN(x0, x1, lxg, lxb, xl);
  LN(y0, y1, lyg, lyb, yl);
  LN(z0, z1, lzg, lzb, zl);
}

// ---------------------------------------------------------------------------
// 4) Fused flash attention with product-of-scores softmax.
//    grid = (B*NH, N/128); block = (32,8): each wave owns 16 query rows.
//    K/Vx/Vy chunks double-buffered in LDS via async global->LDS copies
//    (ASYNCcnt) so the next chunk's DMA overlaps the current chunk's WMMAs.
// ---------------------------------------------------------------------------
__global__ void __launch_bounds__(256) attn_kernel(
    const float* __restrict__ QVx, const float* __restrict__ QVy,
    const float* __restrict__ Km, float* __restrict__ AOx,
    float* __restrict__ AOy) {
  constexpr int MC = 128;
  __shared__ __align__(16) float Ks[2][MC * 8];
  __shared__ __align__(16) float Vxs[2][MC * 8];
  __shared__ __align__(16) float Vys[2][MC * 8];
  __shared__ float Pst[8][16 * 17];  // stride 17: bank-conflict-free A re-read

  const int bh = blockIdx.x;
  const int b = bh >> 3;
  const int h = bh & 7;
  const int wv = threadIdx.y;
  const int lane = threadIdx.x;
  const int r = lane & 15;
  const int kh = (lane >> 4) << 1;
  const int hi8 = (lane >> 4) << 3;
  const float scale = 0.35355339059327373f;  // hd^-0.5, folded into both q's

  const int q0 = blockIdx.y * 128 + wv * 16;
  const int qrow = b * Nn + q0 + r;

  v2f qa[2], qb[2];
#pragma unroll
  for (int s = 0; s < 2; ++s)
#pragma unroll
    for (int e = 0; e < 2; ++e) {
      qa[s][e] = QVx[(size_t)qrow * 128 + h * 8 + s * 4 + kh + e] * scale;
      qb[s][e] = QVy[(size_t)qrow * 128 + h * 8 + s * 4 + kh + e] * scale;
    }

  v8f accx = {}, accy = {};
  float mrun[8], lrun[8];
#pragma unroll
  for (int i = 0; i < 8; ++i) { mrun[i] = -1e30f; lrun[i] = 0.f; }

  const int tid = wv * 32 + lane;
  const int lj = tid >> 1;        // 0..127: key row staged by this thread
  const int lh = (tid & 1) * 4;   // which float4 of the 8-float row

  // stage one 128-key chunk of K / Vx / Vy into LDS buffer (ch & 1)
  auto issue = [&](int ch) {
    const int bufi = ch & 1;
    const int grow = b * Nn + ch * MC + lj;
#ifdef USE_ASYNC_LDS
    async_b128(&Km [(size_t)grow * 64 + h * 8 + lh],       &Ks [bufi][lj * 8 + lh]);
    async_b128(&QVx[(size_t)grow * 128 + 64 + h * 8 + lh], &Vxs[bufi][lj * 8 + lh]);
    async_b128(&QVy[(size_t)grow * 128 + 64 + h * 8 + lh], &Vys[bufi][lj * 8 + lh]);
#else
    *(float4*)&Ks [bufi][lj * 8 + lh] = *(const float4*)&Km [(size_t)grow * 64 + h * 8 + lh];
    *(float4*)&Vxs[bufi][lj * 8 + lh] = *(const float4*)&QVx[(size_t)grow * 128 + 64 + h * 8 + lh];
    *(float4*)&Vys[bufi][lj * 8 + lh] = *(const float4*)&QVy[(size_t)grow * 128 + 64 + h * 8 + lh];
#endif
  };

  issue(0);

  for (int ch = 0; ch < Nn / MC; ++ch) {
#ifdef USE_ASYNC_LDS
    __builtin_amdgcn_s_wait_asynccnt(0);  // this wave's chunk-ch copies landed
#endif
    __syncthreads();                      // => whole buffer (ch&1) is valid
    if (ch + 1 < Nn / MC) issue(ch + 1);  // overlap next DMA with compute

    const float* ks = Ks [ch & 1];
    const float* vx = Vxs[ch & 1];
    const float* vy = Vys[ch & 1];

    for (int t = 0; t < MC / 16; ++t) {
      // scores: s1 = q_x K^T * scale, s2 = q_y K^T * scale  (K-dim = 8)
      v8f s1 = {}, s2 = {};
#pragma unroll
      for (int s = 0; s < 2; ++s) {
        v2f bk;
        bk[0] = ks[(t * 16 + r) * 8 + s * 4 + kh];
        bk[1] = ks[(t * 16 + r) * 8 + s * 4 + kh + 1];
        s1 = wmma4(qa[s], bk, s1);
        s2 = wmma4(qb[s], bk, s2);
      }
      // online softmax of the elementwise product s1*s2
#pragma unroll
      for (int i = 0; i < 8; ++i) {
        float sc = s1[i] * s2[i];
        float mn = fmaxf(mrun[i], half_max16(sc));
        float pe = __expf(sc - mn);
        float ps = half_sum16(pe);
        float corr = __expf(mrun[i] - mn);
        mrun[i] = mn;
        lrun[i] = lrun[i] * corr + ps;
        accx[i] *= corr;
        accy[i] *= corr;
        Pst[wv][(i + hi8) * 17 + r] = pe;  // C-layout -> LDS
      }
      // P(16x16) @ V(16x8, cols 8..15 zero-padded), K-dim = 16 -> 4 steps
#pragma unroll
      for (int kk = 0; kk < 4; ++kk) {
        v2f pa, bx, by;
#pragma unroll
        for (int e = 0; e < 2; ++e) {
          int kloc = kk * 4 + kh + e;
          pa[e] = Pst[wv][r * 17 + kloc];  // A-layout re-read
          bx[e] = (r < 8) ? vx[(t * 16 + kloc) * 8 + r] : 0.f;
          by[e] = (r < 8) ? vy[(t * 16 + kloc) * 8 + r] : 0.f;
        }
        accx = wmma4(pa, bx, accx);
        accy = wmma4(pa, by, accy);
      }
    }
  }

  if (r < 8) {
#pragma unroll
    for (int i = 0; i < 8; ++i) {
      int n = b * Nn + q0 + i + hi8;
      float inv = 1.f / lrun[i];
      AOx[(size_t)n * 64 + h * 8 + r] = accx[i] * inv;
      AOy[(size_t)n * 64 + h * 8 + r] = accy[i] * inv;
    }
  }
}

// ---------------------------------------------------------------------------
// 5) proj GEMM + residual, scattered to NCHW
// ---------------------------------------------------------------------------
__global__ void __launch_bounds__(256) gemm_proj_kernel(
    const float* __restrict__ A, const float* __restrict__ W,
    const float* __restrict__ bias, const float* __restrict__ res,
    float* __restrict__ out) {
  const int K = 64, Nc = 64;
  const int wid = blockIdx.x * blockDim.y + threadIdx.y;
  const int mt = wid / (Nc >> 4), nt = wid % (Nc >> 4);
  const int lane = threadIdx.x;
  const int r = lane & 15;
  const int kh = (lane >> 4) << 1;
  const int hi8 = (lane >> 4) << 3;
  const float* Ar = A + (size_t)(mt * 16 + r) * K;
  const float* Wr = W + (size_t)(nt * 16 + r) * K;
  v8f acc = {};
  for (int k0 = 0; k0 < K; k0 += 4) {
    v2f a, b;
    a[0] = Ar[k0 + kh]; a[1] = Ar[k0 + kh + 1];
    b[0] = Wr[k0 + kh]; b[1] = Wr[k0 + kh + 1];
    acc = wmma4(a, b, acc);
  }
  int col = nt * 16 + r;
#pragma unroll
  for (int i = 0; i < 8; ++i) {
    int gg = mt * 16 + i + hi8;
    int b2 = gg / Nn, pp = gg % Nn;
    out[(size_t)(b2 * 64 + col) * Nn + pp] =
        acc[i] + bias[col] + res[(size_t)gg * 64 + col];
  }
}

// ---------------------------------------------------------------------------
// 6) Implicit-GEMM conv (ks=3 pad=1 or ks=1 pad=0) with fused BN/ReLU/residual
//    NT = Cout/16; one wave computes 16 pixels x Cout.
//    k-order: k = tap*Cin + ci  (tap = dy*3+dx); weight at co*K + ci*taps + tap
// ---------------------------------------------------------------------------
template <int NT>
__global__ void __launch_bounds__(256) conv_wmma_kernel(
    const float* __restrict__ in, const float* __restrict__ wt,
    const float* __restrict__ cbias, const float* __restrict__ bng,
    const float* __restrict__ bnb, const float* __restrict__ bnm,
    const float* __restrict__ bnv, const float* __restrict__ res,
    float* __restrict__ out, int Cin, int cshift, int ks, int relu) {
  const int Cout = NT * 16;
  const int taps = ks * ks;
  const int K = Cin * taps;
  const int pad = ks >> 1;

  const int wid = blockIdx.x * blockDim.y + threadIdx.y;
  const int lane = threadIdx.x;
  const int r = lane & 15;
  const int kh = (lane >> 4) << 1;
  const int hi8 = (lane >> 4) << 3;

  const int g = wid * 16 + r;
  const int bb = g / Nn;
  const int p = g % Nn;
  const int h0 = p / Wn;
  const int w0 = p % Wn;

  v8f acc[NT];
#pragma unroll
  for (int nt = 0; nt < NT; ++nt) acc[nt] = {};

  for (int k0 = 0; k0 < K; k0 += 4) {
    v2f a;
    int wo[2];
#pragma unroll
    for (int e = 0; e < 2; ++e) {
      int k = k0 + kh + e;
      int ci = k & (Cin - 1);
      int tap = k >> cshift;
      int y2 = h0 + tap / 3 - pad;
      int x2 = w0 + tap % 3 - pad;
      float v = 0.f;
      if ((unsigned)y2 < (unsigned)Hn && (unsigned)x2 < (unsigned)Wn)
        v = in[(size_t)(bb * Cin + ci) * Nn + y2 * Wn + x2];
      a[e] = v;
      wo[e] = ci * taps + tap;
    }
#pragma unroll
    for (int nt = 0; nt < NT; ++nt) {
      int co = nt * 16 + r;
      v2f bv;
      bv[0] = wt[(size_t)co * K + wo[0]];
      bv[1] = wt[(size_t)co * K + wo[1]];
      acc[nt] = wmma4(a, bv, acc[nt]);
    }
  }

#pragma unroll
  for (int nt = 0; nt < NT; ++nt) {
    int co = nt * 16 + r;
    float cb = cbias ? cbias[co] : 0.f;
    float sc = 1.f, sh = 0.f;
    if (bng) { sc = bng[co] * rsqrtf(bnv[co] + LEPS); sh = bnb[co] - bnm[co] * sc; }
#pragma unroll
    for (int i = 0; i < 8; ++i) {
      int gg = wid * 16 + i + hi8;
      int b2 = gg / Nn, pp = gg % Nn;
      float v = acc[nt][i] + cb;
      if (bng) v = v * sc + sh;
      if (relu) v = fmaxf(v, 0.f);
      size_t oi = (size_t)(b2 * Cout + co) * Nn + pp;
      if (res) v += res[oi];
      out[oi] = v;
    }
  }
}

// ---------------------------------------------------------------------------
// 7) cat = [xo+yo, xo*yo]; fbn = bn2(cat)
// ---------------------------------------------------------------------------
__global__ void __launch_bounds__(256) catbn_kernel(
    const float* __restrict__ xo, const float* __restrict__ yo,
    const float* g, const float* be, const float* m, const float* v,
    float* __restrict__ cat, float* __restrict__ fbn) {
  int idx = blockIdx.x * 256 + threadIdx.x;  // over B*64*N
  if (idx >= Bn * 64 * Nn) return;
  int pp = idx % Nn;
  int t = idx / Nn;
  int c = t % 64;
  int b = t / 64;
  float a = xo[idx], o = yo[idx];
  float s = a + o, pr = a * o;
  size_t i0 = (size_t)(b * 128 + c) * Nn + pp;
  size_t i1 = (size_t)(b * 128 + 64 + c) * Nn + pp;
  cat[i0] = s;
  cat[i1] = pr;
  float s0 = g[c] * rsqrtf(v[c] + LEPS);
  float s1 = g[64 + c] * rsqrtf(v[64 + c] + LEPS);
  fbn[i0] = (s - m[c]) * s0 + be[c];
  fbn[i1] = (pr - m[64 + c]) * s1 + be[64 + c];
}

// ---------------------------------------------------------------------------
extern "C" void kernel_launch(void* const* d_in, const int* in_sizes, int n_in,
                              void* d_out, int out_size, void* d_ws,
                              size_t ws_size, hipStream_t stream) {
  (void)in_sizes; (void)n_in; (void)out_size; (void)ws_size;
  int a = 0;
  const float* x = (const float*)d_in[a++];
  const float* y = (const float*)d_in[a++];
  const float* pconv_w = (const float*)d_in[a++];
  const float* pconv_b = (const float*)d_in[a++];
  const float* bnd_g = (const float*)d_in[a++];
  const float* bnd_b = (const float*)d_in[a++];
  const float* bnd_m = (const float*)d_in[a++];
  const float* bnd_v = (const float*)d_in[a++];
  const float* lnx_g = (const float*)d_in[a++];
  const float* lnx_b = (const float*)d_in[a++];
  const float* lny_g = (const float*)d_in[a++];
  const float* lny_b = (const float*)d_in[a++];
  const float* lnz_g = (const float*)d_in[a++];
  const float* lnz_b = (const float*)d_in[a++];
  const float* k_w = (const float*)d_in[a++];
  const float* k_b = (const float*)d_in[a++];
  const float* qv_w = (const float*)d_in[a++];
  const float* qv_b = (const float*)d_in[a++];
  const float* proj_w = (const float*)d_in[a++];
  const float* proj_b = (const float*)d_in[a++];
  const float* c2w1 = (const float*)d_in[a++];
  const float* c2b1 = (const float*)d_in[a++];
  const float* c2bn1_g = (const float*)d_in[a++];
  const float* c2bn1_b = (const float*)d_in[a++];
  const float* c2bn1_m = (const float*)d_in[a++];
  const float* c2bn1_v = (const float*)d_in[a++];
  const float* c2w2 = (const float*)d_in[a++];
  const float* c2b2 = (const float*)d_in[a++];
  const float* c2bn2_g = (const float*)d_in[a++];
  const float* c2bn2_b = (const float*)d_in[a++];
  const float* c2bn2_m = (const float*)d_in[a++];
  const float* c2bn2_v = (const float*)d_in[a++];
  const float* bn2_g = (const float*)d_in[a++];
  const float* bn2_b = (const float*)d_in[a++];
  const float* bn2_m = (const float*)d_in[a++];
  const float* bn2_v = (const float*)d_in[a++];
  const float* f1w = (const float*)d_in[a++];
  const float* fbn1_g = (const float*)d_in[a++];
  const float* fbn1_b = (const float*)d_in[a++];
  const float* fbn1_m = (const float*)d_in[a++];
  const float* fbn1_v = (const float*)d_in[a++];
  const float* f2w = (const float*)d_in[a++];
  const float* fbn2_g = (const float*)d_in[a++];
  const float* fbn2_b = (const float*)d_in[a++];
  const float* fbn2_m = (const float*)d_in[a++];
  const float* fbn2_v = (const float*)d_in[a++];
  const float* f3w = (const float*)d_in[a++];
  const float* f3b = (const float*)d_in[a++];

  float* ws = (float*)d_ws;
  size_t off = 0;
  auto F = [&](size_t n) {
    float* p = ws + off;
    off += (n + 63) & ~(size_t)63;
    return p;
  };
  float* X2 = F((size_t)Rn * 128);
  float* XY = F((size_t)Rn * 64);
  float* sc1 = F((size_t)Rn * 64);
  float* sd1 = F((size_t)Rn * 64);
  float* xl = F((size_t)Rn * 64);
  float* yl = F((size_t)Rn * 64);
  float* zl = F((size_t)Rn * 64);
  float* QVx = F((size_t)Rn * 128);
  float* QVy = F((size_t)Rn * 128);
  float* Kmm = F((size_t)Rn * 64);
  float* AOx = F((size_t)Rn * 64);
  float* AOy = F((size_t)Rn * 64);
  float* xo = F((size_t)Rn * 64);
  float* yo = F((size_t)Rn * 64);
  float* t1 = F((size_t)Rn * 64);
  float* xo2 = F((size_t)Rn * 64);
  float* yo2 = F((size_t)Rn * 64);
  float* cat = F((size_t)Rn * 128);
  float* fbn = F((size_t)Rn * 128);
  float* g1 = F((size_t)Rn * 128);
  float* g2 = F((size_t)Rn * 128);

  dim3 blk(32, 8);
  const int tilesM = Rn / 16;  // 288

  pack_kernel<<<Rn * 128 / 256, 256, 0, stream>>>(x, y, X2);
  // pconv: [R,128] @ [64,128]^T
  gemm_rm_kernel<<<tilesM * 4 / 8, blk, 0, stream>>>(X2, pconv_w, pconv_b, XY, 128, 64);
  bnln_kernel<<<Rn / 8, blk, 0, stream>>>(X2, XY, bnd_g, bnd_b, bnd_m, bnd_v,
                                          lnx_g, lnx_b, lny_g, lny_b, lnz_g,
                                          lnz_b, sc1, sd1, xl, yl, zl);
  gemm_rm_kernel<<<tilesM * 8 / 8, blk, 0, stream>>>(xl, qv_w, qv_b, QVx, 64, 128);
  gemm_rm_kernel<<<tilesM * 8 / 8, blk, 0, stream>>>(yl, qv_w, qv_b, QVy, 64, 128);
  gemm_rm_kernel<<<tilesM * 4 / 8, blk, 0, stream>>>(zl, k_w, k_b, Kmm, 64, 64);

  attn_kernel<<<dim3(Bn * 8, Nn / 128), blk, 0, stream>>>(QVx, QVy, Kmm, AOx, AOy);

  gemm_proj_kernel<<<tilesM * 4 / 8, blk, 0, stream>>>(AOx, proj_w, proj_b, sc1, xo);
  gemm_proj_kernel<<<tilesM * 4 / 8, blk, 0, stream>>>(AOy, proj_w, proj_b, sd1, yo);

  const int cblk = tilesM / 8;  // 36
  // conv2 block, branch x
  conv_wmma_kernel<4><<<cblk, blk, 0, stream>>>(xo, c2w1, c2b1, c2bn1_g, c2bn1_b,
      c2bn1_m, c2bn1_v, nullptr, t1, 64, 6, 3, 1);
  conv_wmma_kernel<4><<<cblk, blk, 0, stream>>>(t1, c2w2, c2b2, c2bn2_g, c2bn2_b,
      c2bn2_m, c2bn2_v, xo, xo2, 64, 6, 3, 1);
  // conv2 block, branch y (shared weights)
  conv_wmma_kernel<4><<<cblk, blk, 0, stream>>>(yo, c2w1, c2b1, c2bn1_g, c2bn1_b,
      c2bn1_m, c2bn1_v, nullptr, t1, 64, 6, 3, 1);
  conv_wmma_kernel<4><<<cblk, blk, 0, stream>>>(t1, c2w2, c2b2, c2bn2_g, c2bn2_b,
      c2bn2_m, c2bn2_v, yo, yo2, 64, 6, 3, 1);

  catbn_kernel<<<Rn * 64 / 256, 256, 0, stream>>>(xo2, yo2, bn2_g, bn2_b, bn2_m,
                                                  bn2_v, cat, fbn);

  conv_wmma_kernel<8><<<cblk, blk, 0, stream>>>(fbn, f1w, nullptr, fbn1_g, fbn1_b,
      fbn1_m, fbn1_v, nullptr, g1, 128, 7, 3, 1);
  conv_wmma_kernel<8><<<cblk, blk, 0, stream>>>(g1, f2w, nullptr, fbn2_g, fbn2_b,
      fbn2_m, fbn2_v, nullptr, g2, 128, 7, 3, 1);
  // f3: 1x1 conv + bias, residual add of cat, no bn/relu -> final output
  conv_wmma_kernel<8><<<cblk, blk, 0, stream>>>(g2, f3w, f3b, nullptr, nullptr,
      nullptr, nullptr, cat, (float*)d_out, 128, 7, 1, 0);
}